// HyperboloidAggregation_68272800137557
// MI455X (gfx1250) — compile-verified
//
#include <hip/hip_runtime.h>
#include <hip/hip_bf16.h>

typedef __attribute__((ext_vector_type(2))) float v2f;
typedef __attribute__((ext_vector_type(8))) float v8f;

#define EPSV 1e-6f

__global__ void HyperZero_kernel(float* __restrict__ p, int n) {
  int i = blockIdx.x * blockDim.x + threadIdx.x;
  if (i < n) p[i] = 0.0f;
}

// One wave handles 32 edges. Row-mean of messages (Ex8) done via
// V_WMMA_F32_16X16X4_F32 against an all-ones B (full f32 precision):
// two accumulating WMMAs (K=0..3, K=4..7) per 16-edge tile, two tiles per wave.
__global__ __launch_bounds__(256) void HyperEdge_kernel(
    const float* __restrict__ vertices,
    const int*   __restrict__ edges,
    const float* __restrict__ messages,
    float* __restrict__ sums,
    float* __restrict__ cnt,
    long long E) {
  const int lane = threadIdx.x & 31;
  const long long wid = (long long)blockIdx.x * (blockDim.x >> 5) + (threadIdx.x >> 5);
  const long long e0 = wid << 5;
  if (e0 >= E) return;  // wave-uniform

  // Lane -> edge permutation (swap middle octets) so that after the WMMA,
  // each lane holds its own edge's row sum in accumulator element (lane&7).
  const int octet = lane >> 3;
  const int perm  = (octet == 1) ? (lane + 8) : ((octet == 2) ? (lane - 8) : lane);
  const long long eg = e0 + perm;

  float weight = 0.0f;
  if (e0 + 32 <= E) {  // wave-uniform: full block -> EXEC all ones for WMMA
    // 16x4 f32 A layout: lanes 0-15 hold row=lane, K=0..1; lanes 16-31 K=2..3.
    const int rowA = lane & 15;
    const int kb   = (lane < 16) ? 0 : 2;
    const float* m1 = messages + (e0 + rowA) * 8 + kb;   // edges e0   .. e0+15
    const float* m2 = m1 + 16 * 8;                       // edges e0+16.. e0+31
    v2f a1lo = *(const v2f*)(m1);
    v2f a1hi = *(const v2f*)(m1 + 4);
    v2f a2lo = *(const v2f*)(m2);
    v2f a2hi = *(const v2f*)(m2 + 4);
    v2f ones; ones[0] = 1.0f; ones[1] = 1.0f;   // B == all ones -> D[m][n] = rowsum(m)

    v8f c1 = {};
    c1 = __builtin_amdgcn_wmma_f32_16x16x4_f32(false, a1lo, false, ones, (short)0, c1, false, false);
    c1 = __builtin_amdgcn_wmma_f32_16x16x4_f32(false, a1hi, false, ones, (short)0, c1, false, false);
    v8f c2 = {};
    c2 = __builtin_amdgcn_wmma_f32_16x16x4_f32(false, a2lo, false, ones, (short)0, c2, false, false);
    c2 = __builtin_amdgcn_wmma_f32_16x16x4_f32(false, a2hi, false, ones, (short)0, c2, false, false);

    const bool useD2 = (lane >> 3) & 1;       // octets 1,3 read tile #2
    v8f d = useD2 ? c2 : c1;
    const int r = lane & 7;
    float s = d[0];
    if (r == 1) s = d[1];
    if (r == 2) s = d[2];
    if (r == 3) s = d[3];
    if (r == 4) s = d[4];
    if (r == 5) s = d[5];
    if (r == 6) s = d[6];
    if (r == 7) s = d[7];
    weight = s * 0.125f;
  } else if (eg < E) {  // scalar tail (never taken for E % 32 == 0)
    float s = 0.0f;
#pragma unroll
    for (int k = 0; k < 8; ++k) s += messages[eg * 8 + k];
    weight = s * 0.125f;
  }

  if (eg < E) {
    const int vs = edges[2 * eg + 0];
    const int vd = edges[2 * eg + 1];
    const float x0 = vertices[3 * vs + 0], x1 = vertices[3 * vs + 1], x2 = vertices[3 * vs + 2];
    const float y0 = vertices[3 * vd + 0], y1 = vertices[3 * vd + 1], y2 = vertices[3 * vd + 2];
    const float t  = -x0 * y0 + x1 * y1 + x2 * y2;      // Minkowski <x,y>
    const float v0 = y0 + x0 * t;
    const float v1 = y1 + x1 * t;
    const float v2 = y2 + x2 * t;
    const float vn   = sqrtf(-v0 * v0 + v1 * v1 + v2 * v2 + EPSV);  // = sqrt(t^2-1+eps)
    const float dist = acoshf(-t);
    const float sc   = dist * weight / vn;
    atomicAdd(&sums[3 * vs + 0], v0 * sc);
    atomicAdd(&sums[3 * vs + 1], v1 * sc);
    atomicAdd(&sums[3 * vs + 2], v2 * sc);
    atomicAdd(&cnt[vs], 1.0f);
  }
}

__global__ void HyperVertex_kernel(const float* __restrict__ vertices,
                                   const float* __restrict__ sums,
                                   const float* __restrict__ cnt,
                                   float* __restrict__ out, int N) {
  int i = blockIdx.x * blockDim.x + threadIdx.x;
  if (i >= N) return;
  const float c   = cnt[i];
  const float inv = 1.0f / fmaxf(c, 1.0f);
  float a0 = 0.0f, a1 = 0.0f, a2 = 0.0f;
  if (c > 0.0f) {
    a0 = sums[3 * i + 0] * inv;
    a1 = sums[3 * i + 1] * inv;
    a2 = sums[3 * i + 2] * inv;
  }
  const float an = sqrtf(-a0 * a0 + a1 * a1 + a2 * a2 + EPSV);
  const float T  = an;                 // sqrt(K) == 1
  const float e  = expf(T);
  const float ei = 1.0f / e;
  const float ch = 0.5f * (e + ei);
  const float sh = 0.5f * (e - ei);
  const float f  = sh / an;
  out[3 * i + 0] = ch * vertices[3 * i + 0] + f * a0;
  out[3 * i + 1] = ch * vertices[3 * i + 1] + f * a1;
  out[3 * i + 2] = ch * vertices[3 * i + 2] + f * a2;
}

extern "C" void kernel_launch(void* const* d_in, const int* in_sizes, int n_in,
                              void* d_out, int out_size, void* d_ws, size_t ws_size,
                              hipStream_t stream) {
  const float* vertices = (const float*)d_in[0];   // (N,3) f32
  const int*   edges    = (const int*)d_in[1];     // (E,2) i32
  const float* messages = (const float*)d_in[2];   // (E,8) f32
  float* out = (float*)d_out;                      // (N,3) f32

  const int       N = in_sizes[0] / 3;
  const long long E = (long long)in_sizes[1] / 2;

  float* sums = (float*)d_ws;               // 3*N floats
  float* cnt  = sums + (size_t)3 * N;       // N floats

  const int zn = 4 * N;
  HyperZero_kernel<<<(zn + 255) / 256, 256, 0, stream>>>(sums, zn);

  const long long waves  = (E + 31) >> 5;
  const long long blocks = (waves + 7) >> 3;   // 8 waves (256 threads) per block
  HyperEdge_kernel<<<dim3((unsigned)blocks), 256, 0, stream>>>(
      vertices, edges, messages, sums, cnt, E);

  HyperVertex_kernel<<<(N + 255) / 256, 256, 0, stream>>>(vertices, sums, cnt, out, N);
}